// GCNEncoder_85882166051006
// MI455X (gfx1250) — compile-verified
//
#include <hip/hip_runtime.h>

typedef __attribute__((ext_vector_type(2))) float v2f;
typedef __attribute__((ext_vector_type(8))) float v8f;

__device__ __forceinline__ void atomAddF32(float* p, float v) {
  __hip_atomic_fetch_add(p, v, __ATOMIC_RELAXED, __HIP_MEMORY_SCOPE_AGENT);
}

// ---------------- degree / normalization ----------------

__global__ void set_ones_kernel(float* __restrict__ deg, int n) {
  int i = blockIdx.x * blockDim.x + threadIdx.x;
  if (i < n) deg[i] = 1.0f;  // self-loop contributes 1 to every degree
}

__global__ void count_deg_kernel(const long long* __restrict__ ei,
                                 float* __restrict__ deg, long long E) {
  long long e = blockIdx.x * (long long)blockDim.x + threadIdx.x;
  if (e < E) {
    int d = (int)ei[E + e];
    atomAddF32(&deg[d], 1.0f);
  }
}

__global__ void rsqrt_kernel(float* __restrict__ dinv, int n) {
  int i = blockIdx.x * blockDim.x + threadIdx.x;
  if (i < n) dinv[i] = rsqrtf(dinv[i]);  // deg >= 1 always (self-loops)
}

// ---------------- fused GEMM + dinv scale (WMMA f32 16x16x4) ----------------
// H[r, :] = dinv[r] * (X[r, :] @ W)     X:[n,CIN] row-major, W:[CIN,COUT] row-major
template <int CIN, int COUT>
__global__ __launch_bounds__(256) void gemm_scale_kernel(
    const float* __restrict__ X, const float* __restrict__ W,
    const float* __restrict__ dinv, float* __restrict__ H, int n) {
  constexpr int BM = 64;              // rows per block
  constexpr int KP = CIN + 4;         // padded LDS row stride (bank-conflict avoidance)
  constexpr int NT = COUT / 16;       // wave tiles along N
  constexpr int WR = 8 / NT;          // wave rows
  constexpr int MT = (BM / 16) / WR;  // 16-row M tiles per wave

  __shared__ __align__(16) float xs[BM * KP];    // X tile
  __shared__ __align__(16) float wt[COUT * KP];  // W transposed: wt[n][k]

  const int tid = threadIdx.x;
  const int m0 = blockIdx.x * BM;

  // stage W transposed (coalesced global reads over columns)
  for (int i = tid; i < CIN * COUT; i += 256) {
    int k = i / COUT, c = i % COUT;
    wt[c * KP + k] = W[i];
  }
  // stage X tile (zero-fill tail rows)
  for (int i = tid; i < BM * CIN; i += 256) {
    int r = i / CIN, k = i % CIN;
    int gr = m0 + r;
    xs[r * KP + k] = (gr < n) ? X[(long long)gr * CIN + k] : 0.0f;
  }
  __syncthreads();

  const int wave = tid >> 5;
  const int lane = tid & 31;
  const int half = lane >> 4;  // K-pair selector per WMMA f32 A/B layout
  const int lm = lane & 15;    // row (A) / column (B) index
  const int nt = wave % NT;
  const int wr = wave / NT;
  const int n0 = nt * 16;

  v8f acc[MT] = {};

  for (int k = 0; k < CIN; k += 4) {
    const int kk = k + 2 * half;  // lanes 0-15: K={k,k+1}; lanes 16-31: K={k+2,k+3}
    v2f b = *(const v2f*)&wt[(n0 + lm) * KP + kk];
#pragma unroll
    for (int j = 0; j < MT; ++j) {
      const int row = (wr * MT + j) * 16 + lm;
      v2f a = *(const v2f*)&xs[row * KP + kk];
      // D = A(16x4) * B(4x16) + C ;  v_wmma_f32_16x16x4_f32
      acc[j] = __builtin_amdgcn_wmma_f32_16x16x4_f32(
          /*neg_a=*/false, a, /*neg_b=*/false, b,
          /*c_mod=*/(short)0, acc[j], /*reuse_a=*/false, /*reuse_b=*/false);
    }
  }

  // epilogue: scale rows by dinv and store.
  if (m0 + BM <= n) {
    // fast path (all blocks except the last): branch-free stores
#pragma unroll
    for (int j = 0; j < MT; ++j) {
      const int baseRow = m0 + (wr * MT + j) * 16 + 8 * half;
#pragma unroll
      for (int v = 0; v < 8; ++v) {
        const int row = baseRow + v;  // C/D layout: VGPR v -> M = v + 8*half
        H[(long long)row * COUT + n0 + lm] = dinv[row] * acc[j][v];
      }
    }
  } else {
    // tail block only
#pragma unroll
    for (int j = 0; j < MT; ++j) {
      const int baseRow = m0 + (wr * MT + j) * 16 + 8 * half;
#pragma unroll
      for (int v = 0; v < 8; ++v) {
        const int row = baseRow + v;
        if (row < n) H[(long long)row * COUT + n0 + lm] = dinv[row] * acc[j][v];
      }
    }
  }
}

// ---------------- aggregation ----------------

__global__ void copy4_kernel(const float4* __restrict__ src, float4* __restrict__ dst,
                             long long n4) {
  long long i = blockIdx.x * (long long)blockDim.x + threadIdx.x;
  if (i < n4) dst[i] = src[i];
}

// C/4 threads per edge: gather h'[src] row (float4/lane), atomic-add into acc[dst]
template <int C>
__global__ void scatter_add_kernel(const float* __restrict__ H,
                                   const long long* __restrict__ ei,
                                   float* __restrict__ acc, long long E) {
  constexpr int CH = C / 4;
  long long idx = blockIdx.x * (long long)blockDim.x + threadIdx.x;
  long long e = idx / CH;
  if (e >= E) return;
  int c4 = (int)(idx - e * CH) * 4;
  int s = (int)ei[e];
  int d = (int)ei[E + e];
  const float4 v = *(const float4*)&H[(long long)s * C + c4];
  float* p = &acc[(long long)d * C + c4];
  atomAddF32(p + 0, v.x);
  atomAddF32(p + 1, v.y);
  atomAddF32(p + 2, v.z);
  atomAddF32(p + 3, v.w);
}

template <int C, bool RELU>
__global__ void finalize_kernel(const float* __restrict__ acc,
                                const float* __restrict__ dinv,
                                const float* __restrict__ bias,
                                float* __restrict__ out, int n) {
  long long idx = blockIdx.x * (long long)blockDim.x + threadIdx.x;
  if (idx >= (long long)n * C) return;
  int i = (int)(idx / C);
  int c = (int)(idx % C);
  float v = dinv[i] * acc[idx] + bias[c];
  if (RELU) v = fmaxf(v, 0.0f);
  out[idx] = v;
}

// ---------------- driver ----------------

extern "C" void kernel_launch(void* const* d_in, const int* in_sizes, int n_in,
                              void* d_out, int out_size, void* d_ws, size_t ws_size,
                              hipStream_t stream) {
  const float* x = (const float*)d_in[0];
  const long long* ei = (const long long*)d_in[1];  // [2, E] int64: src then dst
  const float* W1 = (const float*)d_in[2];
  const float* b1 = (const float*)d_in[3];
  const float* W2 = (const float*)d_in[4];
  const float* b2 = (const float*)d_in[5];

  const int chid = in_sizes[3];                    // 128
  const int cout = in_sizes[5];                    // 64
  const int cin = in_sizes[2] / chid;              // 256
  const int n = in_sizes[0] / cin;                 // 100000
  const long long E = (long long)in_sizes[1] / 2;  // 1.6M

  float* ws = (float*)d_ws;
  float* dinv = ws;                     // n
  float* h1 = dinv + n;                 // n*chid, reused as relu output z
  float* acc1 = h1 + (size_t)n * chid;  // n*chid
  float* h2 = acc1 + (size_t)n * chid;  // n*cout
  float* out = (float*)d_out;           // doubles as layer-2 accumulator

  const int T = 256;

  // symmetric normalization coefficients
  set_ones_kernel<<<(n + T - 1) / T, T, 0, stream>>>(dinv, n);
  count_deg_kernel<<<(int)((E + T - 1) / T), T, 0, stream>>>(ei, dinv, E);
  rsqrt_kernel<<<(n + T - 1) / T, T, 0, stream>>>(dinv, n);

  // ---- layer 1: h1 = dinv .* (x @ W1) ----
  gemm_scale_kernel<256, 128><<<(n + 63) / 64, T, 0, stream>>>(x, W1, dinv, h1, n);
  long long n4a = (long long)n * chid / 4;
  copy4_kernel<<<(int)((n4a + T - 1) / T), T, 0, stream>>>((const float4*)h1,
                                                           (float4*)acc1, n4a);
  long long work1 = E * (chid / 4);
  scatter_add_kernel<128><<<(int)((work1 + T - 1) / T), T, 0, stream>>>(h1, ei, acc1, E);
  long long tot1 = (long long)n * chid;
  finalize_kernel<128, true><<<(int)((tot1 + T - 1) / T), T, 0, stream>>>(acc1, dinv, b1,
                                                                          h1, n);

  // ---- layer 2: h2 = dinv .* (z @ W2) ----
  gemm_scale_kernel<128, 64><<<(n + 63) / 64, T, 0, stream>>>(h1, W2, dinv, h2, n);
  long long n4b = (long long)n * cout / 4;
  copy4_kernel<<<(int)((n4b + T - 1) / T), T, 0, stream>>>((const float4*)h2,
                                                           (float4*)out, n4b);
  long long work2 = E * (cout / 4);
  scatter_add_kernel<64><<<(int)((work2 + T - 1) / T), T, 0, stream>>>(h2, ei, out, E);
  long long tot2 = (long long)n * cout;
  finalize_kernel<64, false><<<(int)((tot2 + T - 1) / T), T, 0, stream>>>(out, dinv, b2,
                                                                          out, n);
}